// InsideLoss3DTriMesh_9758165696610
// MI455X (gfx1250) — compile-verified
//
#include <hip/hip_runtime.h>
#include <hip/hip_bf16.h>

typedef float v2f __attribute__((ext_vector_type(2)));
typedef float v8f __attribute__((ext_vector_type(8)));
typedef int   v4i __attribute__((ext_vector_type(4)));

#define EPS_LOSS 0.01f
#define NBATCH   2
#define NPTS     10     // interpolation points per face kept by the simplex filter
#define CHUNK    2000   // shape points staged per LDS buffer (2000*16B = 32KB; x2 bufs = 64KB)

#define LDS_AS __attribute__((address_space(3)))
#define GLB_AS __attribute__((address_space(1)))

// w rows in numpy a-major order, filtered to (1-a-b) >= 0, t = {0,1/3,2/3,1}
__constant__ float c_w[NPTS][3] = {
    {0.0f,          0.0f,          1.0f},
    {0.0f,          0.33333334f,   0.6666666f},
    {0.0f,          0.6666667f,    0.33333331f},
    {0.0f,          1.0f,          0.0f},
    {0.33333334f,   0.0f,          0.6666666f},
    {0.33333334f,   0.33333334f,   0.33333331f},
    {0.33333334f,   0.6666667f,   -0.0f},
    {0.6666667f,    0.0f,          0.33333331f},
    {0.6666667f,    0.33333334f,  -0.0f},
    {1.0f,          0.0f,          0.0f},
};

__device__ __forceinline__ void async_copy_b128(const float4* g, float4* l) {
#if __has_builtin(__builtin_amdgcn_global_load_async_to_lds_b128)
    __builtin_amdgcn_global_load_async_to_lds_b128(
        (GLB_AS v4i*)g, (LDS_AS v4i*)l, /*offset=*/0, /*cpol=*/0);
#else
    *l = *g;  // synchronous fallback
#endif
}

__device__ __forceinline__ void wait_async0() {
#if __has_builtin(__builtin_amdgcn_s_wait_asynccnt)
    __builtin_amdgcn_s_wait_asynccnt(0);
#else
    asm volatile("s_wait_asynccnt 0x0" ::: "memory");
#endif
}

// Kernel 1: q[b, f*NPTS+p, :] = sum_k w[p][k] * cage_v[b, cage_f[b,f,k], :]
__global__ void build_q_kernel(const float* __restrict__ cage_v,
                               const int* __restrict__ cage_f,
                               float* __restrict__ qbuf,
                               int M, int F) {
    int tid = blockIdx.x * blockDim.x + threadIdx.x;
    int total = NBATCH * F * NPTS;
    if (tid >= total) return;
    int b = tid / (F * NPTS);
    int rem = tid - b * (F * NPTS);
    int f = rem / NPTS;
    int p = rem - f * NPTS;

    int i0 = cage_f[((size_t)b * F + f) * 3 + 0];
    int i1 = cage_f[((size_t)b * F + f) * 3 + 1];
    int i2 = cage_f[((size_t)b * F + f) * 3 + 2];
    const float* v0 = cage_v + ((size_t)b * M + i0) * 3;
    const float* v1 = cage_v + ((size_t)b * M + i1) * 3;
    const float* v2 = cage_v + ((size_t)b * M + i2) * 3;
    float w0 = c_w[p][0], w1 = c_w[p][1], w2 = c_w[p][2];

    float* out = qbuf + (size_t)tid * 3;
    out[0] = w0 * v0[0] + w1 * v1[0] + w2 * v2[0];
    out[1] = w0 * v0[1] + w1 * v1[1] + w2 * v2[1];
    out[2] = w0 * v0[2] + w1 * v1[2] + w2 * v2[2];
}

// Kernel 2: pack shape point n as float4 {x, y, z, |n|^2}
__global__ void pack_shape_kernel(const float* __restrict__ shape,
                                  float* __restrict__ psh, int totalN) {
    int tid = blockIdx.x * blockDim.x + threadIdx.x;
    if (tid >= totalN) return;
    float x = shape[(size_t)tid * 3 + 0];
    float y = shape[(size_t)tid * 3 + 1];
    float z = shape[(size_t)tid * 3 + 2];
    ((float4*)psh)[tid] = make_float4(x, y, z, x * x + y * y + z * z);
}

// Kernel 3: WMMA nearest-neighbor + inside-loss partials.
// 8 waves/block, one 16-query tile per wave; shape chunks staged to LDS with
// double-buffered async global->LDS copies (ASYNCcnt) and consumed via ds loads.
// Score(m,n) = |n|^2 - 2 q_m . p_n via v_wmma_f32_16x16x4_f32 with
// A = [-2qx, -2qy | -2qz, 1] and B rows = [x; y | z; |n|^2].
__global__ __launch_bounds__(256) void nn_loss_kernel(
    const float* __restrict__ qbuf,      // [B][S][3]
    const float* __restrict__ psh,       // [B][N][4] packed
    const float* __restrict__ shape_vn,  // [B][N][3]
    float* __restrict__ partials,        // [numWaves]
    int S, int N, int numTiles) {
    __shared__ float4 sbuf[2][CHUNK];

    int wid  = (blockIdx.x * blockDim.x + threadIdx.x) >> 5;
    int lane = threadIdx.x & 31;
    // All waves must reach every barrier: clamp instead of returning.
    bool active = (wid < numTiles);
    int widc = active ? wid : (numTiles - 1);

    int tilesPerBatch = S >> 4;
    int b    = widc / tilesPerBatch;
    int tile = widc - b * tilesPerBatch;
    int row  = lane & 15;
    int s0   = tile << 4;

    // ---- A operand (16x4 f32): lanes 0-15 hold K=0,1; lanes 16-31 hold K=2,3
    const float* qp = qbuf + ((size_t)b * S + s0 + row) * 3;
    float qx = qp[0], qy = qp[1], qz = qp[2];
    v2f a;
    if (lane < 16) { a.x = -2.0f * qx; a.y = -2.0f * qy; }
    else           { a.x = -2.0f * qz; a.y = 1.0f; }

    float minv[8];
    int   mini[8];
#pragma unroll
    for (int r = 0; r < 8; ++r) { minv[r] = 3.4e38f; mini[r] = 0; }

    const float4* pb4 = (const float4*)psh + (size_t)b * N;
    int colOff = (lane < 16) ? 0 : 2;  // lanes 0-15: (x,y); lanes 16-31: (z,|n|^2)
    v8f czero = {};

    int nChunks = (N + CHUNK - 1) / CHUNK;

    // Prologue: stage chunk 0.
    {
        int nPts = min(CHUNK, N);
        for (int i = threadIdx.x; i < CHUNK; i += blockDim.x) {
            if (i < nPts) async_copy_b128(pb4 + i, &sbuf[0][i]);
            else          sbuf[0][i] = make_float4(0.f, 0.f, 0.f, 3.0e38f);
        }
    }
    wait_async0();
    __syncthreads();

    for (int k = 0; k < nChunks; ++k) {
        int cur = k & 1;
        int n0  = k * CHUNK;

        // Prefetch next chunk into the other buffer while we compute.
        if (k + 1 < nChunks) {
            int nb   = n0 + CHUNK;
            int nPts = min(CHUNK, N - nb);
            for (int i = threadIdx.x; i < CHUNK; i += blockDim.x) {
                if (i < nPts) async_copy_b128(pb4 + nb + i, &sbuf[1 - cur][i]);
                else          sbuf[1 - cur][i] = make_float4(0.f, 0.f, 0.f, 3.0e38f);
            }
        }

        // Consume current chunk: CHUNK/16 WMMAs per wave, B operand loads
        // software-pipelined one tile ahead so ds_load latency overlaps the
        // WMMA + argmin of the previous tile.
        const float* sb = (const float*)&sbuf[cur][0];
        v2f bm;
        {
            const float* sp = sb + (size_t)row * 4 + colOff;
            bm.x = sp[0];
            bm.y = sp[1];
        }
        int c0 = 0;
#pragma unroll 2
        for (; c0 + 16 < CHUNK; c0 += 16) {
            const float* sn = sb + (size_t)(c0 + 16 + row) * 4 + colOff;
            v2f bmn;
            bmn.x = sn[0];   // ds_load_b64 for next tile (in flight during WMMA)
            bmn.y = sn[1];

            v8f d = __builtin_amdgcn_wmma_f32_16x16x4_f32(
                false, a, false, bm, (short)0, czero, false, false);

            int col = n0 + c0 + row;
#pragma unroll
            for (int r = 0; r < 8; ++r) {
                float dv = d[r];
                if (dv < minv[r]) { minv[r] = dv; mini[r] = col; }
            }
            bm = bmn;
        }
        {   // epilogue: last tile of the chunk (no further load)
            v8f d = __builtin_amdgcn_wmma_f32_16x16x4_f32(
                false, a, false, bm, (short)0, czero, false, false);
            int col = n0 + c0 + row;
#pragma unroll
            for (int r = 0; r < 8; ++r) {
                float dv = d[r];
                if (dv < minv[r]) { minv[r] = dv; mini[r] = col; }
            }
        }

        wait_async0();     // next chunk's async copies complete (per-wave)
        __syncthreads();   // publish LDS across waves; guard buffer reuse
    }

    // Reduce (min,argmin) across the 16-lane halves (C layout: lanes 0-15 rows 0-7,
    // lanes 16-31 rows 8-15). xor masks 1..8 stay within each half on wave32.
#pragma unroll
    for (int r = 0; r < 8; ++r) {
#pragma unroll
        for (int off = 8; off >= 1; off >>= 1) {
            float ov = __shfl_xor(minv[r], off, 32);
            int   oi = __shfl_xor(mini[r], off, 32);
            if (ov < minv[r] || (ov == minv[r] && oi < mini[r])) {
                minv[r] = ov; mini[r] = oi;
            }
        }
    }

    float acc = 0.0f;
    if ((lane & 15) == 0) {  // lane 0 -> rows 0..7, lane 16 -> rows 8..15
        int rbase = (lane >= 16) ? 8 : 0;
#pragma unroll
        for (int r = 0; r < 8; ++r) {
            int srow = s0 + rbase + r;
            const float* qq = qbuf + ((size_t)b * S + srow) * 3;
            int ni = mini[r];
            const float4 pp = pb4[ni];
            const float* nn = shape_vn + ((size_t)b * N + ni) * 3;
            float nx = nn[0], ny = nn[1], nz = nn[2];
            float dx = qq[0] - pp.x - EPS_LOSS * nx;
            float dy = qq[1] - pp.y - EPS_LOSS * ny;
            float dz = qq[2] - pp.z - EPS_LOSS * nz;
            float dot = dx * nx + dy * ny + dz * nz;
            acc += (dot < 0.0f) ? -dot : 0.0f;
        }
    }
    acc += __shfl_xor(acc, 16, 32);
    if (active && lane == 0) partials[wid] = acc;
}

// Kernel 4: deterministic reduction of per-wave partials.
__global__ __launch_bounds__(256) void reduce_kernel(const float* __restrict__ partials,
                                                     int n, float* __restrict__ out,
                                                     float invCount) {
    __shared__ float sm[256];
    float s = 0.0f;
    for (int i = threadIdx.x; i < n; i += 256) s += partials[i];
    sm[threadIdx.x] = s;
    __syncthreads();
    for (int st = 128; st > 0; st >>= 1) {
        if ((int)threadIdx.x < st) sm[threadIdx.x] += sm[threadIdx.x + st];
        __syncthreads();
    }
    if (threadIdx.x == 0) out[0] = sm[0] * invCount;
}

extern "C" void kernel_launch(void* const* d_in, const int* in_sizes, int n_in,
                              void* d_out, int out_size, void* d_ws, size_t ws_size,
                              hipStream_t stream) {
    const float* cage_v   = (const float*)d_in[0];
    const int*   cage_f   = (const int*)d_in[1];
    const float* shape    = (const float*)d_in[2];
    const float* shape_vn = (const float*)d_in[3];

    const int B = NBATCH;
    const int M = in_sizes[0] / (B * 3);   // 642
    const int F = in_sizes[1] / (B * 3);   // 1000
    const int N = in_sizes[2] / (B * 3);   // 20000
    const int S = F * NPTS;                // 10000 (multiple of 16)

    // Workspace layout
    char* ws = (char*)d_ws;
    float* qbuf = (float*)ws;                             // B*S*3 floats
    size_t qBytes = (size_t)B * S * 3 * sizeof(float);
    qBytes = (qBytes + 15) & ~(size_t)15;
    float* psh = (float*)(ws + qBytes);                   // B*N*4 floats (float4)
    size_t pBytes = (size_t)B * N * 4 * sizeof(float);
    float* partials = (float*)(ws + qBytes + pBytes);     // numTiles floats

    int numTiles = B * (S / 16);

    {
        int total = B * F * NPTS;
        build_q_kernel<<<(total + 255) / 256, 256, 0, stream>>>(cage_v, cage_f, qbuf, M, F);
    }
    {
        int total = B * N;
        pack_shape_kernel<<<(total + 255) / 256, 256, 0, stream>>>(shape, psh, total);
    }
    {
        int wavesPerBlock = 256 / 32;
        int blocks = (numTiles + wavesPerBlock - 1) / wavesPerBlock;
        nn_loss_kernel<<<blocks, 256, 0, stream>>>(qbuf, psh, shape_vn, partials, S, N, numTiles);
    }
    {
        float invCount = 1.0f / ((float)B * (float)S);
        reduce_kernel<<<1, 256, 0, stream>>>(partials, numTiles, (float*)d_out, invCount);
    }
}